// Upsample_82154134438530
// MI455X (gfx1250) — compile-verified
//
#include <hip/hip_runtime.h>
#include <stdint.h>

// ---------------------------------------------------------------------------
// StyleGAN2 upfirdn2d upsample x2, kernel [1,3,3,1] (separable, 4x4).
// Memory-bound: ~320MB traffic @ 23.3 TB/s => ~14us floor. No matmul structure
// => use the CDNA5 async-DMA path (GLOBAL_LOAD_ASYNC_TO_LDS_B128, ASYNCcnt)
// for input staging + nontemporal 128-bit streaming stores for the output.
// ---------------------------------------------------------------------------

#define AS1 __attribute__((address_space(1)))
#define AS3 __attribute__((address_space(3)))

typedef float v4f __attribute__((ext_vector_type(4)));
typedef int   v4i __attribute__((vector_size(16)));   // matches builtin param type

__device__ __forceinline__ void async_copy_b128(AS3 float* lds_dst, const float* gsrc) {
#if __has_builtin(__builtin_amdgcn_global_load_async_to_lds_b128)
  __builtin_amdgcn_global_load_async_to_lds_b128(
      (AS1 v4i*)(void*)gsrc, (AS3 v4i*)lds_dst, 0, 0);
#else
  uint32_t loff = (uint32_t)(uintptr_t)lds_dst;   // AS3 value == LDS byte offset
  asm volatile("global_load_async_to_lds_b128 %0, %1, off"
               :: "v"(loff), "v"((const void*)gsrc)
               : "memory");
#endif
}

__device__ __forceinline__ void wait_async0() {
#if __has_builtin(__builtin_amdgcn_s_wait_asynccnt)
  __builtin_amdgcn_s_wait_asynccnt(0);
#else
  asm volatile("s_wait_asynccnt 0" ::: "memory");
#endif
}

static constexpr int H = 128, W = 128;   // input plane
static constexpr int R = 32;             // input rows per strip
static constexpr int LROWS = R + 2;      // + top/bottom halo
static constexpr int NF4 = LROWS * W / 4; // 1088 float4 staging slots

__global__ __launch_bounds__(256)
void Upsample_82154134438530_kernel(const float* __restrict__ x,
                                    const float* __restrict__ kk,   // 4x4 kernel
                                    float* __restrict__ out) {
  __shared__ float smem[LROWS * W];                 // 17408 B

  const int tid   = threadIdx.x;
  const int plane = blockIdx.x >> 2;                // N*C planes
  const int strip = blockIdx.x & 3;                 // 4 strips of 32 rows
  const int r0    = strip * R;

  const float* xp = x + (size_t)plane * (H * W);
  float*       op = out + (size_t)plane * (size_t)(4 * H * W);

  // ---- stage strip (+halo rows) into LDS via async DMA --------------------
  AS3 float* lbase = (AS3 float*)smem;
#pragma unroll
  for (int s = 0; s < 5; ++s) {
    int f4 = s * 256 + tid;
    if (f4 < NF4) {
      int row = f4 >> 5;                 // 32 float4 per row
      int gr  = r0 - 1 + row;            // global input row (-1..128)
      if (gr >= 0 && gr < H) {
        async_copy_b128(lbase + f4 * 4, xp + gr * W + (f4 & 31) * 4);
      } else {
        v4f z = {0.f, 0.f, 0.f, 0.f};    // zero the halo rows outside the image
        *(v4f*)(&smem[f4 * 4]) = z;
      }
    }
  }
  wait_async0();
  __syncthreads();

  // ---- compute: each thread emits one float4 of output per iteration -----
  const int tx = tid & 63;               // 64 float4 across 256 output cols
  const int ty = tid >> 6;               // 4 output rows at a time
  const int a  = ty & 1;                 // output row parity (wave-uniform)

  // true convolution: out[o] = sum_s K[s] * up[o+1-s]; parity selects taps.
  const int syP = a ? 2 : 1;             // tap on row j
  const int syS = a ? 0 : 3;             // tap on row j+1 (odd) / j-1 (even)
  const float wPP0 = kk[syP * 4 + 1], wPS0 = kk[syP * 4 + 3];   // even col
  const float wPP1 = kk[syP * 4 + 2], wPS1 = kk[syP * 4 + 0];   // odd  col
  const float wSP0 = kk[syS * 4 + 1], wSS0 = kk[syS * 4 + 3];
  const float wSP1 = kk[syS * 4 + 2], wSS1 = kk[syS * 4 + 0];

  const int i0 = 2 * tx;                 // input cols covered by this thread
  const int i1 = i0 + 1;

#pragma unroll 4
  for (int it = 0; it < 16; ++it) {
    const int oy = strip * 64 + it * 4 + ty;   // global output row
    const int j  = oy >> 1;                    // primary input row
    const int rP = j - r0 + 1;                 // LDS row of j
    const int rS = a ? rP + 1 : rP - 1;        // LDS row of secondary tap
    const float* rowP = &smem[rP * W];
    const float* rowS = &smem[rS * W];

    const float pA = rowP[i0], pB = rowP[i1];
    const float sA = rowS[i0], sB = rowS[i1];
    const float pL = (i0 > 0)     ? rowP[i0 - 1] : 0.f;
    const float sL = (i0 > 0)     ? rowS[i0 - 1] : 0.f;
    const float pR = (i1 < W - 1) ? rowP[i1 + 1] : 0.f;
    const float sR = (i1 < W - 1) ? rowS[i1 + 1] : 0.f;

    v4f o;
    o.x = wPP0 * pA + wPS0 * pL + wSP0 * sA + wSS0 * sL;  // ox = 4tx   (even)
    o.y = wPP1 * pA + wPS1 * pB + wSP1 * sA + wSS1 * sB;  // ox = 4tx+1 (odd)
    o.z = wPP0 * pB + wPS0 * pA + wSP0 * sB + wSS0 * sA;  // ox = 4tx+2 (even)
    o.w = wPP1 * pB + wPS1 * pR + wSP1 * sB + wSS1 * sR;  // ox = 4tx+3 (odd)

    // streaming store: don't pollute L2 with the 256MB write-once output
    __builtin_nontemporal_store(o, (v4f*)(op + (size_t)oy * 256 + tx * 4));
  }
}

extern "C" void kernel_launch(void* const* d_in, const int* in_sizes, int n_in,
                              void* d_out, int out_size, void* d_ws, size_t ws_size,
                              hipStream_t stream) {
  (void)n_in; (void)out_size; (void)d_ws; (void)ws_size;
  const float* x  = (const float*)d_in[0];   // (16,64,128,128) fp32
  const float* k  = (const float*)d_in[1];   // (4,4) fp32
  float*       o  = (float*)d_out;           // (16,64,256,256) fp32

  const int planes = in_sizes[0] / (H * W);  // 1024
  dim3 grid(planes * (H / R));               // 4096 blocks, 1 strip each
  Upsample_82154134438530_kernel<<<grid, 256, 0, stream>>>(x, k, o);
}